// Model_visual_sp_28003186770673
// MI455X (gfx1250) — compile-verified
//
#include <hip/hip_runtime.h>
#include <hip/hip_bf16.h>

#define USER_N   100000
#define ITEM_N   50000
#define NODES    (USER_N + ITEM_N)      // 150000
#define LATDIM   64
#define IMG_DIM  1024
#define E_ADJ    3000000
#define E_IMG    1000000
#define EPSV     1e-12f

typedef float v2f __attribute__((ext_vector_type(2)));
typedef float v8f __attribute__((ext_vector_type(8)));

// ---------------------------------------------------------------------------
// Fused:  feats_n = L2_normalize_rows( A[ITEM,1024] @ W[1024,64] + b )
// One workgroup = 16 output rows. 4 waves, one 16x16 N-tile each.
// Uses V_WMMA_F32_16X16X4_F32 (full f32 precision; GEMM is memory-bound).
// ---------------------------------------------------------------------------
__global__ __launch_bounds__(128)
void gemm_bias_norm_kernel(const float* __restrict__ A,     // [ITEM,1024]
                           const float* __restrict__ W,     // [1024,64]
                           const float* __restrict__ bias,  // [64]
                           float* __restrict__ out)         // [ITEM,64]
{
    const int m0   = blockIdx.x * 16;
    const int wave = threadIdx.x >> 5;      // 0..3 -> N-tile
    const int lane = threadIdx.x & 31;
    const int half = lane >> 4;             // 0: lanes 0-15, 1: lanes 16-31
    const int l16  = lane & 15;
    const int n    = wave * 16 + l16;       // global column 0..63

    __shared__ float norm2[16];
    if (threadIdx.x < 16) norm2[threadIdx.x] = 0.0f;
    __syncthreads();

    // Bias depends only on the column -> same value in all 8 C VGPRs.
    const float bn = bias[n];
    v8f c;
#pragma unroll
    for (int r = 0; r < 8; ++r) c[r] = bn;

    // A fragment: lane l holds row (l%16), K = 2*(l/16)+{0,1}  (contiguous pair)
    const float* __restrict__ arow = A + (size_t)(m0 + l16) * IMG_DIM;

    for (int k0 = 0; k0 < IMG_DIM; k0 += 4) {
        const int k = k0 + 2 * half;
        v2f a;
        a.x = arow[k];
        a.y = arow[k + 1];
        // B fragment: lane l holds col (l%16), K = 2*(l/16)+v
        v2f b;
        b.x = W[(size_t)k * LATDIM + n];
        b.y = W[(size_t)(k + 1) * LATDIM + n];
        c = __builtin_amdgcn_wmma_f32_16x16x4_f32(
                /*neg_a=*/false, a, /*neg_b=*/false, b,
                /*c_mod=*/(short)0, c, /*reuse_a=*/false, /*reuse_b=*/false);
    }

    // Row sum-of-squares: reduce each C VGPR across the 16 lanes of its half.
    float ss[8];
#pragma unroll
    for (int r = 0; r < 8; ++r) {
        float s = c[r] * c[r];
        s += __shfl_xor(s, 1, 32);
        s += __shfl_xor(s, 2, 32);
        s += __shfl_xor(s, 4, 32);
        s += __shfl_xor(s, 8, 32);
        ss[r] = s;
    }
    if (l16 == 0) {
#pragma unroll
        for (int r = 0; r < 8; ++r)
            atomicAdd(&norm2[r + 8 * half], ss[r]);   // ds_add_f32 across 4 waves
    }
    __syncthreads();

    // Scale by 1/max(||row||, eps) and store.
#pragma unroll
    for (int r = 0; r < 8; ++r) {
        const int row   = r + 8 * half;
        const float nrm = sqrtf(norm2[row]);
        const float scl = 1.0f / fmaxf(nrm, EPSV);
        out[(size_t)(m0 + row) * LATDIM + n] = c[r] * scl;
    }
}

// ---------------------------------------------------------------------------
// Edge-parallel SpMM scatter:  out[r] += v * src(c)   (64-wide rows)
// src(c) = c < split ? srcA[c] : srcB[c - split]
// 4 threads per edge, 16 dims each (float4 gathers, f32 atomics -> L2).
// ---------------------------------------------------------------------------
__global__ __launch_bounds__(256)
void spmm_scatter_kernel(const int* __restrict__ rows,
                         const int* __restrict__ cols,
                         const float* __restrict__ vals, int nnz,
                         const float* __restrict__ srcA,
                         const float* __restrict__ srcB, int split,
                         float* __restrict__ out)
{
    const long long t = (long long)blockIdx.x * blockDim.x + threadIdx.x;
    const long long e = t >> 2;
    if (e >= nnz) return;
    const int q = (int)(t & 3);

    const int   r = rows[e];
    const int   ci = cols[e];
    const float v = vals[e];

    const float* __restrict__ src =
        (ci < split) ? (srcA + (size_t)ci * LATDIM)
                     : (srcB + (size_t)(ci - split) * LATDIM);

    const float4* __restrict__ s4 = (const float4*)(src + q * 16);
    float* __restrict__ o = out + (size_t)r * LATDIM + q * 16;

#pragma unroll
    for (int i = 0; i < 4; ++i) {
        const float4 x = s4[i];
        atomicAdd(o + 4 * i + 0, v * x.x);
        atomicAdd(o + 4 * i + 1, v * x.y);
        atomicAdd(o + 4 * i + 2, v * x.z);
        atomicAdd(o + 4 * i + 3, v * x.w);
    }
}

// ---------------------------------------------------------------------------
// t = a + b + c ;  acc = t ; c (in-place) = t   (RIS_LAMBDA == 1.0)
// ---------------------------------------------------------------------------
__global__ __launch_bounds__(256)
void combine_kernel(const float* __restrict__ a, const float* __restrict__ b,
                    float* __restrict__ c_io, float* __restrict__ acc, size_t n)
{
    const size_t i = (size_t)blockIdx.x * blockDim.x + threadIdx.x;
    if (i < n) {
        const float t = a[i] + b[i] + c_io[i];
        acc[i]  = t;
        c_io[i] = t;
    }
}

__global__ __launch_bounds__(256)
void accumulate_kernel(float* __restrict__ acc, const float* __restrict__ x, size_t n)
{
    const size_t i = (size_t)blockIdx.x * blockDim.x + threadIdx.x;
    if (i < n) acc[i] += x[i];
}

// ---------------------------------------------------------------------------
extern "C" void kernel_launch(void* const* d_in, const int* in_sizes, int n_in,
                              void* d_out, int out_size, void* d_ws, size_t ws_size,
                              hipStream_t stream)
{
    const int*   adj_rows = (const int*)  d_in[0];
    const int*   adj_cols = (const int*)  d_in[1];
    const float* adj_vals = (const float*)d_in[2];
    const int*   img_rows = (const int*)  d_in[3];
    const int*   img_cols = (const int*)  d_in[4];
    const float* img_vals = (const float*)d_in[5];
    const float* img_emb  = (const float*)d_in[6];   // [ITEM, 1024]
    const float* uEmb     = (const float*)d_in[7];   // [USER, 64]
    const float* iEmb     = (const float*)d_in[8];   // [ITEM, 64]
    const float* W_trans  = (const float*)d_in[9];   // [1024, 64]
    const float* b_trans  = (const float*)d_in[10];  // [64]

    float* acc = (float*)d_out;                      // [NODES, 64]

    // Workspace layout (floats): feats_n + 3 ping-pong [NODES,64] buffers.
    float* ws      = (float*)d_ws;
    float* feats_n = ws;                                         // ITEM*64
    float* bufA    = feats_n + (size_t)ITEM_N * LATDIM;          // embeds_image
    float* bufB    = bufA    + (size_t)NODES  * LATDIM;          // embeds_image_
    float* bufC    = bufB    + (size_t)NODES  * LATDIM;          // img_adj / cur

    const size_t nElem   = (size_t)NODES * LATDIM;               // 9.6M
    const size_t nBytes  = nElem * sizeof(float);
    const int    ewBlk   = 256;
    const dim3   gridEl((unsigned)((nElem + ewBlk - 1) / ewBlk));
    const dim3   gridAdj((unsigned)(((long long)E_ADJ * 4 + ewBlk - 1) / ewBlk));
    const dim3   gridImg((unsigned)(((long long)E_IMG * 4 + ewBlk - 1) / ewBlk));

    // 1) feats_n = normalize(img_emb @ W + b)   -- fused WMMA GEMM
    gemm_bias_norm_kernel<<<ITEM_N / 16, 128, 0, stream>>>(img_emb, W_trans, b_trans, feats_n);

    // 2) bufC = img_adj @ concat(u, i)
    hipMemsetAsync(bufC, 0, nBytes, stream);
    spmm_scatter_kernel<<<gridImg, ewBlk, 0, stream>>>(
        img_rows, img_cols, img_vals, E_IMG, uEmb, iEmb, USER_N, bufC);

    // 3) bufA = adj @ concat(u, feats_n)
    hipMemsetAsync(bufA, 0, nBytes, stream);
    spmm_scatter_kernel<<<gridAdj, ewBlk, 0, stream>>>(
        adj_rows, adj_cols, adj_vals, E_ADJ, uEmb, feats_n, USER_N, bufA);

    // 4) bufB = adj @ concat(bufA[:USER], i)
    hipMemsetAsync(bufB, 0, nBytes, stream);
    spmm_scatter_kernel<<<gridAdj, ewBlk, 0, stream>>>(
        adj_rows, adj_cols, adj_vals, E_ADJ, bufA, iEmb, USER_N, bufB);

    // 5) cur = bufA + bufB + bufC ; acc = cur  (cur stored in bufC)
    combine_kernel<<<gridEl, ewBlk, 0, stream>>>(bufA, bufB, bufC, acc, nElem);

    // 6) GCN layer 1: bufB = adj @ cur(bufC); acc += bufB
    hipMemsetAsync(bufB, 0, nBytes, stream);
    spmm_scatter_kernel<<<gridAdj, ewBlk, 0, stream>>>(
        adj_rows, adj_cols, adj_vals, E_ADJ, bufC, bufC, NODES, bufB);
    accumulate_kernel<<<gridEl, ewBlk, 0, stream>>>(acc, bufB, nElem);

    // 7) GCN layer 2: bufC = adj @ bufB; acc += bufC
    hipMemsetAsync(bufC, 0, nBytes, stream);
    spmm_scatter_kernel<<<gridAdj, ewBlk, 0, stream>>>(
        adj_rows, adj_cols, adj_vals, E_ADJ, bufB, bufB, NODES, bufC);
    accumulate_kernel<<<gridEl, ewBlk, 0, stream>>>(acc, bufC, nElem);

    (void)in_sizes; (void)n_in; (void)out_size; (void)ws_size;
}